// FNO1d_56667798503924
// MI455X (gfx1250) — compile-verified
//
#include <hip/hip_runtime.h>

typedef __attribute__((ext_vector_type(2))) float v2f;
typedef __attribute__((ext_vector_type(4))) float v4f;
typedef __attribute__((ext_vector_type(8))) float v8f;

#define PI_D 3.14159265358979323846

static __device__ __forceinline__ v8f wmma_f32(v2f a, v2f b, v8f c) {
  // D = A(16x4) * B(4x16) + C, f32 in / f32 accumulate
  return __builtin_amdgcn_wmma_f32_16x16x4_f32(false, a, false, b, (short)0, c, false, false);
}

// wave id made explicitly wave-uniform so loop bounds/trip counts are scalar
static __device__ __forceinline__ int wave_id() {
  return __builtin_amdgcn_readfirstlane(blockIdx.x * (blockDim.x >> 5) + (threadIdx.x >> 5));
}

#define BB 64          // batch
#define CC 64          // channels (D_V)
#define NT 8192        // sequence length
#define NPAD 8200      // padded DFT length (NT + X_PAD)
#define NST 8208       // storage stride (multiple of 16)
#define MODES_ 64
#define KK2 128        // 2*MODES (re | im)
#define NLAYERS 4
#define ROWS 4096      // BB*CC
#define KSF 2050       // NPAD/4 K-steps, forward DFT
#define KSLICES 8
#define KSI 32         // KK2/4 K-steps, inverse DFT
#define NTILES_ 513    // NST/16 column tiles

// ---------------- basis / weight packing (fragment lane order) ----------------

// forward basis BF[n][col]: col<64 -> cos(2pi*col*n/NPAD), col>=64 -> -sin(2pi*(col-64)*n/NPAD)
// packed as float2 per (kstep s, coltile t, lane):  bf[((s*8+t)*32+lane)*2]
__global__ void k_basisF(float* __restrict__ bf) {
  int t = blockIdx.x * blockDim.x + threadIdx.x;
  if (t >= KSF * 8 * 32) return;
  int lane = t & 31, tile = (t >> 5) & 7, s = t >> 8;
  int laneLo = lane & 15, h = lane >> 4;
  int col = tile * 16 + laneLo;
  int mode = (col < MODES_) ? col : col - MODES_;
  int n0 = s * 4 + 2 * h;
  float v[2];
#pragma unroll
  for (int j = 0; j < 2; j++) {
    long long r = ((long long)mode * (long long)(n0 + j)) % NPAD;
    double th = (double)r * (2.0 * PI_D / (double)NPAD);
    v[j] = (col < MODES_) ? (float)cos(th) : (float)(-sin(th));
  }
  bf[2 * t + 0] = v[0];
  bf[2 * t + 1] = v[1];
}

// inverse basis BI[k2][n]: k2<64 -> c_k*cos(2pi*k*n/N), k2>=64 -> -(2/N)*sin(2pi*k*n/N)
// c_0 = 1/N, c_k = 2/N.  packed per (coltile nt, kstep s, lane): bi[((nt*32+s)*32+lane)*2]
__global__ void k_basisI(float* __restrict__ bi) {
  int t = blockIdx.x * blockDim.x + threadIdx.x;
  if (t >= NTILES_ * KSI * 32) return;
  int lane = t & 31, s = (t >> 5) & 31, tile = t >> 10;
  int laneLo = lane & 15, h = lane >> 4;
  int n = tile * 16 + laneLo;
  int k0 = s * 4 + 2 * h;
  float v[2];
#pragma unroll
  for (int j = 0; j < 2; j++) {
    int kd = k0 + j;
    float val = 0.0f;
    if (n < NPAD) {
      if (kd < MODES_) {
        double c = ((kd == 0) ? 1.0 : 2.0) / (double)NPAD;
        long long r = ((long long)kd * (long long)n) % NPAD;
        val = (float)(c * cos((double)r * (2.0 * PI_D / (double)NPAD)));
      } else {
        int k = kd - MODES_;
        if (k != 0) {
          long long r = ((long long)k * (long long)n) % NPAD;
          val = (float)(-(2.0 / (double)NPAD) * sin((double)r * (2.0 * PI_D / (double)NPAD)));
        }
      }
    }
    v[j] = val;
  }
  bi[2 * t + 0] = v[0];
  bi[2 * t + 1] = v[1];
}

// pack fourier weights fw[L][i][o][k][2] into B-fragment order:
// float4 {wr(i0),wr(i0+1),wi(i0),wi(i0+1)} at fp[(((l*64+k)*16+s)*4+ot)*32+lane]
__global__ void k_packFw(const float* __restrict__ fw, float* __restrict__ fp) {
  int t = blockIdx.x * blockDim.x + threadIdx.x;
  if (t >= NLAYERS * MODES_ * 16 * 4 * 32) return;
  int lane = t & 31, ot = (t >> 5) & 3, s = (t >> 7) & 15, k = (t >> 11) & 63, l = t >> 17;
  int laneLo = lane & 15, h = lane >> 4;
  int o = ot * 16 + laneLo;
  int i0 = s * 4 + 2 * h;
  size_t b0 = ((((size_t)l * 64 + i0) * 64 + o) * 64 + k) * 2;
  size_t b1 = ((((size_t)l * 64 + i0 + 1) * 64 + o) * 64 + k) * 2;
  v4f w;
  w[0] = fw[b0];     w[1] = fw[b1];
  w[2] = fw[b0 + 1]; w[3] = fw[b1 + 1];
  ((v4f*)fp)[t] = w;
}

// ---------------- lift: x[b][c][n] = v*pw0 + grid*pw1 + pb (zero in pad) ------

__global__ void k_lift(const float* __restrict__ v, const float* __restrict__ grid,
                       const float* __restrict__ pw, const float* __restrict__ pb,
                       float* __restrict__ x) {
  size_t t = (size_t)blockIdx.x * blockDim.x + threadIdx.x;
  if (t >= (size_t)ROWS * NST) return;
  int n = (int)(t % NST);
  int row = (int)(t / NST);
  int c = row & 63, b = row >> 6;
  float val = 0.0f;
  if (n < NT) val = v[(size_t)b * NT + n] * pw[c] + grid[n] * pw[64 + c] + pb[c];
  x[t] = val;
}

__global__ void k_zero(float* __restrict__ p, int count) {
  int t = blockIdx.x * blockDim.x + threadIdx.x;
  if (t < count) p[t] = 0.0f;
}

// ---------------- forward truncated DFT (split-K, WMMA f32) -------------------
// Y[row][col], row = b*64+c (4096), col = 0..63 re, 64..127 im

__global__ __launch_bounds__(256) void k_dft(const float* __restrict__ x,
                                             const float* __restrict__ bf,
                                             float* __restrict__ Y) {
  int lane = threadIdx.x & 31;
  int wid = wave_id();
  int rt = wid >> 3, ks = wid & 7;
  int laneLo = lane & 15, h = lane >> 4;
  const v4f* xrow = (const v4f*)(x + (size_t)(rt * 16 + laneLo) * NST);
  const v2f* bfv = (const v2f*)bf;
  v8f acc[8] = {};
  int s0 = ks * 257;
  int s1 = s0 + 257; if (s1 > KSF) s1 = KSF;        // scalar bounds -> scalar loop
  for (int s = s0; s < s1; ++s) {
    v4f xa = xrow[s];
    v2f a; a.x = h ? xa[2] : xa[0]; a.y = h ? xa[3] : xa[1];
    const v2f* bp = bfv + (size_t)s * 256 + lane;
#pragma unroll
    for (int tl = 0; tl < 8; tl++) acc[tl] = wmma_f32(a, bp[tl * 32], acc[tl]);
  }
  int rowD = rt * 16 + 8 * h;
#pragma unroll
  for (int tl = 0; tl < 8; tl++)
#pragma unroll
    for (int e = 0; e < 8; e++)
      unsafeAtomicAdd(&Y[(size_t)(rowD + e) * KK2 + tl * 16 + laneLo], acc[tl][e]);
}

// ---------------- per-mode complex channel mixing (WMMA f32) ------------------
// out_r = Xr*Wr - Xi*Wi ; out_i = Xi*Wr + Xr*Wi

__global__ __launch_bounds__(256) void k_modemix(const float* __restrict__ Y,
                                                 const float* __restrict__ fp,
                                                 float* __restrict__ Ym, int l) {
  int lane = threadIdx.x & 31;
  int wid = wave_id();
  int k = wid >> 4, bt = (wid >> 2) & 3, ot = wid & 3;
  int laneLo = lane & 15, h = lane >> 4;
  int bA = bt * 16 + laneLo;
  v8f accR = {}, accI = {};
  const v4f* wp = ((const v4f*)fp) + (size_t)(l * 64 + k) * 2048 + ot * 32 + lane;
#pragma unroll 4
  for (int s = 0; s < 16; s++) {
    int i0 = s * 4 + 2 * h;
    const float* y0 = Y + (size_t)(bA * 64 + i0) * KK2;
    const float* y1 = Y + (size_t)(bA * 64 + i0 + 1) * KK2;
    v2f ar; ar.x = y0[k];      ar.y = y1[k];
    v2f ai; ai.x = y0[64 + k]; ai.y = y1[64 + k];
    v4f w = wp[s * 128];
    v2f br;  br.x = w[0];  br.y = w[1];
    v2f bi;  bi.x = w[2];  bi.y = w[3];
    v2f nbi; nbi.x = -w[2]; nbi.y = -w[3];
    accR = wmma_f32(ar, br, accR);
    accR = wmma_f32(ai, nbi, accR);
    accI = wmma_f32(ai, br, accI);
    accI = wmma_f32(ar, bi, accI);
  }
#pragma unroll
  for (int e = 0; e < 8; e++) {
    int b = bt * 16 + e + 8 * h, o = ot * 16 + laneLo;
    Ym[(size_t)(b * 64 + o) * KK2 + k] = accR[e];
    Ym[(size_t)(b * 64 + o) * KK2 + 64 + k] = accI[e];
  }
}

// ---------------- fused inverse DFT + channel mix + bias + relu ---------------
// x_next[b][d][n] = relu( sum_k2 Ym[b,d,k2]*BI[k2,n] + sum_c W[c,d]*x[b,c,n] + wb[d] )

__global__ __launch_bounds__(256) void k_invfused(const float* __restrict__ Ym,
                                                  const float* __restrict__ bi,
                                                  const float* __restrict__ x,
                                                  const float* __restrict__ ww,
                                                  const float* __restrict__ wb,
                                                  float* __restrict__ xn, int relu) {
  int lane = threadIdx.x & 31;
  int wid = wave_id();
  if (wid >= BB * NTILES_) return;
  int b = wid / NTILES_, nt = wid % NTILES_;
  int laneLo = lane & 15, h = lane >> 4;
  int n = nt * 16 + laneLo;
  v8f a1[4] = {}, a2[4] = {};
  // x1: inverse DFT over 128 mode coefficients
  const v2f* bip = (const v2f*)bi + (size_t)nt * 1024 + lane;
#pragma unroll 4
  for (int s = 0; s < KSI; s++) {
    v2f bfrag = bip[s * 32];
    int k0 = s * 4 + 2 * h;
#pragma unroll
    for (int dt = 0; dt < 4; dt++) {
      const float* yr = Ym + (size_t)(b * 64 + dt * 16 + laneLo) * KK2 + k0;
      v2f a; a.x = yr[0]; a.y = yr[1];
      a1[dt] = wmma_f32(a, bfrag, a1[dt]);
    }
  }
  // x2: channel mixing over 64 input channels
#pragma unroll 4
  for (int s = 0; s < 16; s++) {
    int c0 = s * 4 + 2 * h;
    v2f bfrag;
    bfrag.x = x[(size_t)(b * 64 + c0) * NST + n];
    bfrag.y = x[(size_t)(b * 64 + c0 + 1) * NST + n];
#pragma unroll
    for (int dt = 0; dt < 4; dt++) {
      int d = dt * 16 + laneLo;
      v2f a; a.x = ww[c0 * 64 + d]; a.y = ww[(c0 + 1) * 64 + d];
      a2[dt] = wmma_f32(a, bfrag, a2[dt]);
    }
  }
#pragma unroll
  for (int dt = 0; dt < 4; dt++)
#pragma unroll
    for (int e = 0; e < 8; e++) {
      int d = dt * 16 + e + 8 * h;
      float val = a1[dt][e] + a2[dt][e] + wb[d];
      if (relu) val = fmaxf(val, 0.0f);
      xn[(size_t)(b * 64 + d) * NST + n] = val;
    }
}

// ---------------- final projection: out[b,n] = sum_c x[b,c,n]*qw[c] + qb ------

__global__ void k_project(const float* __restrict__ x, const float* __restrict__ qw,
                          const float* __restrict__ qb, float* __restrict__ out) {
  int t = blockIdx.x * blockDim.x + threadIdx.x;
  if (t >= BB * NT) return;
  int b = t >> 13, n = t & (NT - 1);
  const float* xr = x + (size_t)b * 64 * NST + n;
  float s = qb[0];
#pragma unroll 8
  for (int c = 0; c < 64; c++) s += xr[(size_t)c * NST] * qw[c];
  out[t] = s;
}

// ---------------- host-side launch -------------------------------------------

extern "C" void kernel_launch(void* const* d_in, const int* in_sizes, int n_in,
                              void* d_out, int out_size, void* d_ws, size_t ws_size,
                              hipStream_t stream) {
  const float* v    = (const float*)d_in[0];
  const float* grid = (const float*)d_in[1];
  const float* pw   = (const float*)d_in[2];
  const float* pb   = (const float*)d_in[3];
  const float* fw   = (const float*)d_in[4];
  const float* ww   = (const float*)d_in[5];
  const float* wb   = (const float*)d_in[6];
  const float* qw   = (const float*)d_in[7];
  const float* qb   = (const float*)d_in[8];
  float* out = (float*)d_out;
  float* ws  = (float*)d_ws;

  const size_t XSZ = (size_t)ROWS * NST;            // 33,619,968 floats
  float* xA = ws;
  float* xB = xA + XSZ;
  float* Y  = xB + XSZ;                             // ROWS*KK2
  float* Ym = Y + (size_t)ROWS * KK2;
  float* bf = Ym + (size_t)ROWS * KK2;              // KSF*8*32*2
  float* bi = bf + (size_t)KSF * 8 * 32 * 2;        // NTILES_*KSI*32*2
  float* fp = bi + (size_t)NTILES_ * KSI * 32 * 2;  // NLAYERS*MODES_*16*4*32*4

  k_basisF<<<2050, 256, 0, stream>>>(bf);
  k_basisI<<<2052, 256, 0, stream>>>(bi);
  k_packFw<<<2048, 256, 0, stream>>>(fw, fp);
  k_lift<<<(int)((XSZ + 255) / 256), 256, 0, stream>>>(v, grid, pw, pb, xA);

  float* xc = xA;
  float* xn = xB;
  for (int l = 0; l < NLAYERS; l++) {
    k_zero<<<2048, 256, 0, stream>>>(Y, ROWS * KK2);
    k_dft<<<256, 256, 0, stream>>>(xc, bf, Y);
    k_modemix<<<128, 256, 0, stream>>>(Y, fp, Ym, l);
    k_invfused<<<4104, 256, 0, stream>>>(Ym, bi, xc, ww + (size_t)l * 64 * 64,
                                         wb + (size_t)l * 64, xn, (l < NLAYERS - 1) ? 1 : 0);
    float* tmp = xc; xc = xn; xn = tmp;
  }
  k_project<<<2048, 256, 0, stream>>>(xc, qw, qb, out);
}